// DWLMLayer_67095979099071
// MI455X (gfx1250) — compile-verified
//
#include <hip/hip_runtime.h>
#include <math.h>

typedef float v2f __attribute__((ext_vector_type(2)));
typedef float v8f __attribute__((ext_vector_type(8)));

#define NB      8
#define NDET    8525
#define NCH     80
#define NLVL    5
#define MAXOBJ  100
#define NTOT    (NB * NDET)          // 68200
#define NBIN    (NB * MAXOBJ * NLVL) // 4000

__device__ __forceinline__ int level_of(int n) {
    return (n < 6400) ? 0 : (n < 8000) ? 1 : (n < 8400) ? 2 : (n < 8500) ? 3 : 4;
}

__device__ __forceinline__ float focal_term(float y, float p) {
    p = fminf(fmaxf(p, 1e-7f), 1.0f - 1e-7f);
    float pt = y * p + (1.0f - y) * (1.0f - p);
    float af = 0.75f - 0.5f * y;               // y*0.25 + (1-y)*0.75
    float om = 1.0f - pt;
    return -af * om * om * __logf(pt);
}

__device__ __forceinline__ float giou_loss(float4 p, float4 t) {
    float ap    = (p.x + p.z) * (p.y + p.w);
    float at    = (t.x + t.z) * (t.y + t.w);
    float inter = (fminf(p.x, t.x) + fminf(p.z, t.z)) * (fminf(p.y, t.y) + fminf(p.w, t.w));
    float uni   = ap + at - inter;
    float enc   = (fmaxf(p.x, t.x) + fmaxf(p.z, t.z)) * (fmaxf(p.y, t.y) + fmaxf(p.w, t.w));
    float g     = inter / fmaxf(uni, 1e-7f) - (enc - uni) / fmaxf(enc, 1e-7f);
    return 1.0f - g;
}

// ---------------- kernel 0: zero the scatter bins ----------------
__global__ void k_zero(float* __restrict__ w, int n) {
    int i = blockIdx.x * blockDim.x + threadIdx.x;
    if (i < n) w[i] = 0.0f;
}

// ---------------- kernel 1: fused focal(WMMA reduce) + giou + scatter ----------------
// One wave32 handles 16 detections. A = 16x4 focal terms (det x channel-chunk),
// B = ones -> C rows accumulate the 80-channel sum on the matrix pipe while the
// VALU produces the log-heavy terms. EXEC is all-1s throughout the WMMA loop
// (out-of-range detections are address-clamped, discarded at the atomic stage).
__global__ void k_loss_scatter(const float* __restrict__ cls_pred,
                               const float* __restrict__ loc_pred,
                               const float* __restrict__ cls_tar,
                               const float* __restrict__ loc_tar,
                               const int*   __restrict__ ind_tar,
                               float* __restrict__ sums,
                               float* __restrict__ cnts) {
    const int lane = threadIdx.x & 31;
    const int wave = threadIdx.x >> 5;
    const int base = (blockIdx.x * 8 + wave) * 16;   // first det of this wave
    if (base >= NTOT) return;                        // wave-uniform

    const int half = lane >> 4;                      // 0 -> K=0,1 ; 1 -> K=2,3
    const int mrow = lane & 15;                      // A-matrix row (det)
    int det = base + mrow;
    int dc  = det < NTOT ? det : NTOT - 1;           // clamp loads, keep EXEC full

    const float* pp = cls_pred + (size_t)dc * NCH;
    const float* yy = cls_tar  + (size_t)dc * (NCH + 2);

    v8f acc = {};
    v2f ones; ones.x = 1.0f; ones.y = 1.0f;

#pragma unroll
    for (int s = 0; s < 20; ++s) {                   // 20 * K4 = 80 channels
        const int c0 = s * 4 + half * 2;
        float2 p2 = *(const float2*)(pp + c0);
        float2 y2 = *(const float2*)(yy + c0);
        v2f a;
        a.x = focal_term(y2.x, p2.x);
        a.y = focal_term(y2.y, p2.y);
        acc = __builtin_amdgcn_wmma_f32_16x16x4_f32(
            /*neg_a=*/false, a, /*neg_b=*/false, ones,
            /*c_mod=*/(short)0, acc, /*reuse_a=*/false, /*reuse_b=*/false);
    }

    // C extraction: vgpr r, lanes 0-15 -> M=r ; lanes 16-31 -> M=8+r
    if ((lane & 15) < 8) {
        const int r = lane & 7;
        float cs;
        switch (r) {
            case 0: cs = acc[0]; break;
            case 1: cs = acc[1]; break;
            case 2: cs = acc[2]; break;
            case 3: cs = acc[3]; break;
            case 4: cs = acc[4]; break;
            case 5: cs = acc[5]; break;
            case 6: cs = acc[6]; break;
            default: cs = acc[7]; break;
        }
        const int d = base + ((lane & 16) ? 8 + r : r);
        if (d < NTOT) {
            float4 P = *(const float4*)(loc_pred + (size_t)d * 4);
            float4 T = *(const float4*)(loc_tar  + (size_t)d * 4);
            float total = cs + giou_loss(P, T);
            int b = d / NDET;
            int n = d - b * NDET;
            int l = level_of(n);
            int m = ind_tar[d];
            m = m < 0 ? 0 : (m > MAXOBJ - 1 ? MAXOBJ - 1 : m);
            int off = (b * MAXOBJ + m) * NLVL + l;
            atomicAdd(&sums[off], total);
            atomicAdd(&cnts[off], 1.0f);
        }
    }
}

// ---------------- kernel 2: per-object level-weight target ----------------
__global__ void k_target(const float* __restrict__ sums,
                         const float* __restrict__ cnts,
                         const int*   __restrict__ bcnt,
                         float* __restrict__ targ) {
    const int b = blockIdx.x;
    const int m = threadIdx.x;
    if (m >= MAXOBJ) return;
    const int base = (b * MAXOBJ + m) * NLVL;

    float mean[NLVL];
    float mn = 3.4e38f, mx = -3.4e38f;
#pragma unroll
    for (int l = 0; l < NLVL; ++l) {
        float me = sums[base + l] / fmaxf(1.0f, cnts[base + l]);
        mean[l] = me;
        mn = fminf(mn, me);
        mx = fmaxf(mx, me);
    }
    float denom = mx - mn;
    if (denom == 0.0f) denom = 1.0f;

    float tg[NLVL];
#pragma unroll
    for (int l = 0; l < NLVL; ++l) tg[l] = 1.0f - (mean[l] - mn) / denom;

    // 4th largest of 5 == 2nd smallest (with multiplicity)
    float s1 = 3.4e38f, s2 = 3.4e38f;
#pragma unroll
    for (int l = 0; l < NLVL; ++l) {
        float v = tg[l];
        if (v < s1) { s2 = s1; s1 = v; }
        else if (v < s2) { s2 = v; }
    }
    float valid = (m < bcnt[b]) ? 1.0f : 0.0f;
#pragma unroll
    for (int l = 0; l < NLVL; ++l) {
        float tv = (tg[l] > s2) ? tg[l] : 0.0f;
        targ[base + l] = tv * valid;
    }
}

// ---------------- kernel 3: per-detection gather + select ----------------
__global__ void k_out(const float* __restrict__ cls_tar,
                      const int*   __restrict__ ind_tar,
                      const float* __restrict__ targ,
                      float* __restrict__ out) {
    int d = blockIdx.x * blockDim.x + threadIdx.x;
    if (d >= NTOT) return;
    float flag = cls_tar[(size_t)d * (NCH + 2) + NCH + 1];
    float v = 1.0f;
    if (flag > 0.0f) {
        int b = d / NDET;
        int n = d - b * NDET;
        int m = ind_tar[d];
        m = m < 0 ? 0 : (m > MAXOBJ - 1 ? MAXOBJ - 1 : m);
        v = targ[(b * MAXOBJ + m) * NLVL + level_of(n)];
    }
    out[d] = v;
}

extern "C" void kernel_launch(void* const* d_in, const int* in_sizes, int n_in,
                              void* d_out, int out_size, void* d_ws, size_t ws_size,
                              hipStream_t stream) {
    const float* cls_pred = (const float*)d_in[0];
    const float* loc_pred = (const float*)d_in[1];
    const float* cls_tar  = (const float*)d_in[2];
    const float* loc_tar  = (const float*)d_in[3];
    const int*   ind_tar  = (const int*)d_in[4];
    const int*   bcnt     = (const int*)d_in[5];
    float* out = (float*)d_out;

    float* ws   = (float*)d_ws;
    float* sums = ws;                 // NBIN floats
    float* cnts = ws + NBIN;          // NBIN floats
    float* targ = ws + 2 * NBIN;      // NBIN floats

    // zero scatter bins (sums + counts)
    k_zero<<<(2 * NBIN + 255) / 256, 256, 0, stream>>>(ws, 2 * NBIN);

    // fused loss + WMMA channel reduction + scatter: 16 dets per wave, 8 waves/block
    int waves  = (NTOT + 15) / 16;
    int blocks = (waves + 7) / 8;
    k_loss_scatter<<<blocks, 256, 0, stream>>>(cls_pred, loc_pred, cls_tar, loc_tar,
                                               ind_tar, sums, cnts);

    // per-(batch, object) target weights
    k_target<<<NB, 128, 0, stream>>>(sums, cnts, bcnt, targ);

    // final per-detection gather/select
    k_out<<<(NTOT + 255) / 256, 256, 0, stream>>>(cls_tar, ind_tar, targ, out);
}